// myGATConv_edge_multi_13975823582064
// MI455X (gfx1250) — compile-verified
//
#include <hip/hip_runtime.h>

// Problem constants (fixed by reference): IN=256, H=4, D=64, EF=8, H*D=256
#define IN_F   256
#define HD     256
#define NHEAD  4
#define DHEAD  64
#define NEG_SLOPE 0.2f

typedef float v2f __attribute__((ext_vector_type(2)));
typedef float v8f __attribute__((ext_vector_type(8)));
typedef unsigned int u32x4 __attribute__((ext_vector_type(4)));
typedef int i32x4 __attribute__((ext_vector_type(4)));
typedef int i32x8 __attribute__((ext_vector_type(8)));

// ---------- order-preserving float <-> uint encoding for atomic max ----------
__device__ __forceinline__ unsigned enc_f(float f) {
    unsigned b = __float_as_uint(f);
    return (b & 0x80000000u) ? ~b : (b | 0x80000000u);
}
__device__ __forceinline__ float dec_f(unsigned u) {
    unsigned b = (u & 0x80000000u) ? (u & 0x7FFFFFFFu) : ~u;
    return __uint_as_float(b);
}

// ---------- TDM: 2-D f32 tile (tile_d1 rows x tile_d0 elems) global -> LDS ----------
// D# layout per CDNA5 ISA §8.3/§8.4. Rows beyond tensor_d1 read as zero (OOB fill).
#if __has_builtin(__builtin_amdgcn_tensor_load_to_lds)
#define HAVE_TDM 1
__device__ __forceinline__ void tdm_load_2d_f32_to_lds(
        const float* gptr, unsigned lds_byte_off,
        unsigned tensor_d0, unsigned tensor_d1,
        unsigned tile_d0, unsigned tile_d1, unsigned stride0_elems) {
    unsigned long long ga = (unsigned long long)(const void*)gptr;
    u32x4 g0;
    g0.x = 1u;                                                  // count=1 (valid), user mode
    g0.y = lds_byte_off;                                        // lds_addr (bytes)
    g0.z = (unsigned)ga;                                        // global_addr[31:0]
    g0.w = (unsigned)((ga >> 32) & 0x01FFFFFFull) | (2u << 30); // addr[56:32] | type=2
    i32x8 g1;
    g1[0] = (int)(2u << 16);                                    // data_size=2 (4 bytes)
    g1[1] = (int)((tensor_d0 & 0xFFFFu) << 16);                 // tensor_dim0[15:0]
    g1[2] = (int)(((tensor_d0 >> 16) & 0xFFFFu) |               // tensor_dim0[31:16]
                  ((tensor_d1 & 0xFFFFu) << 16));               // tensor_dim1[15:0]
    g1[3] = (int)(((tensor_d1 >> 16) & 0xFFFFu) |               // tensor_dim1[31:16]
                  ((tile_d0 & 0xFFFFu) << 16));                 // tile_dim0
    g1[4] = (int)(tile_d1 & 0xFFFFu);                           // tile_dim1 (tile_dim2=0)
    g1[5] = (int)stride0_elems;                                 // tensor_dim0_stride[31:0]
    g1[6] = 0;                                                  // stride0[47:32], stride1[15:0]
    g1[7] = 0;                                                  // stride1[47:16]
    i32x4 gz = {0, 0, 0, 0};                                    // groups 2/3 unused (2-D)
#if defined(__clang_major__) && __clang_major__ >= 23
    i32x8 gz8 = {};
    __builtin_amdgcn_tensor_load_to_lds(g0, g1, gz, gz, gz8, 0);
#else
    __builtin_amdgcn_tensor_load_to_lds(g0, g1, gz, gz, 0);
#endif
}
#else
#define HAVE_TDM 0
#endif

// ---------- 0: init accumulators (must run every launch; graph-replay safe) ----------
__global__ __launch_bounds__(256) void init_k(float* out, unsigned* m_enc, float* s,
                                              float* asum, long long nHD, long long nH) {
    long long i = (long long)blockIdx.x * 256 + threadIdx.x;
    if (i < nHD) out[i] = 0.0f;
    if (i < nH) { m_enc[i] = 0u; s[i] = 0.0f; asum[i] = 0.0f; }
}

// ---------- 1: ft = feat @ W  via V_WMMA_F32_16X16X4_F32 ----------
// Block = 256 threads (8 wave32). One block per 16-row strip of feat.
// Strip (16 x 256 f32 = 16KB) DMA'd into LDS by the Tensor Data Mover (wave 0),
// each wave owns two 16-col tiles of W.
__global__ __launch_bounds__(256) void gemm_ft_wmma(const float* __restrict__ feat,
                                                    const float* __restrict__ W,
                                                    float* __restrict__ ft, int N) {
    __shared__ float As[16 * IN_F];
    const int tileM = blockIdx.x * 16;
    const int tid   = threadIdx.x;

#if HAVE_TDM
    if (tid < 32) {   // wave 0 issues the DMA; TDM ignores EXEC
        unsigned remain = (unsigned)(N - tileM);     // rows past this are OOB-zero-filled
        tdm_load_2d_f32_to_lds(feat + (size_t)tileM * IN_F,
                               (unsigned)(unsigned long long)(void*)As,
                               IN_F, remain, IN_F, 16u, IN_F);
        __builtin_amdgcn_s_wait_tensorcnt(0);
    }
    __syncthreads();
#else
    for (int i = tid; i < 16 * IN_F; i += 256) {
        int r = i >> 8;
        int c = i & (IN_F - 1);
        int gr = tileM + r;
        As[i] = (gr < N) ? feat[(size_t)gr * IN_F + c] : 0.0f;
    }
    __syncthreads();
#endif

    const int wave = tid >> 5;     // 0..7
    const int lane = tid & 31;
    const int l15  = lane & 15;
    const int hi   = lane >> 4;    // 0 or 1
    const bool fullTile = (tileM + 16 <= N);   // uniform: clean stores on fast path

    for (int t = 0; t < 2; ++t) {
        const int tileN = (wave * 2 + t) * 16;
        v8f c = {};
        for (int k0 = 0; k0 < IN_F; k0 += 4) {
            const int ka = k0 + 2 * hi;
            // A 16x4 (ISA layout): lane l15 = row M, VGPR0 = K=ka, VGPR1 = K=ka+1
            v2f a;
            a.x = As[l15 * IN_F + ka];
            a.y = As[l15 * IN_F + ka + 1];
            // B 4x16: row K striped across lanes within a VGPR; halves split on hi
            v2f b;
            b.x = W[(size_t)ka * HD + tileN + l15];
            b.y = W[(size_t)(ka + 1) * HD + tileN + l15];
            c = __builtin_amdgcn_wmma_f32_16x16x4_f32(
                    false, a, false, b, (short)0, c, false, false);
        }
        // D layout: VGPR i -> row = tileM + i + 8*hi, col = tileN + l15
        float* ftp = ft + (size_t)(tileM + 8 * hi) * HD + tileN + l15;
        if (fullTile) {
            #pragma unroll
            for (int i = 0; i < 8; ++i) ftp[(size_t)i * HD] = c[i];
        } else {
            #pragma unroll
            for (int i = 0; i < 8; ++i) {
                int row = tileM + i + 8 * hi;
                if (row < N) ft[(size_t)row * HD + tileN + l15] = c[i];
            }
        }
    }
}

// ---------- 2: edge logits e = leaky_relu(sum_f e_feat*attn_e) ----------
__global__ __launch_bounds__(256) void edge_logits_k(const float* __restrict__ e_feat,
                                                     const float* __restrict__ attn_e,
                                                     float* __restrict__ aE, int EH) {
    int i = blockIdx.x * 256 + threadIdx.x;
    if (i >= EH) return;
    int h = i & (NHEAD - 1);
    const float4* ef = (const float4*)e_feat + (size_t)i * 2;   // 8 floats per (e,h)
    const float4* at = (const float4*)attn_e + h * 2;
    float4 p0 = ef[0], p1 = ef[1];
    float4 q0 = at[0], q1 = at[1];
    float ee = p0.x*q0.x + p0.y*q0.y + p0.z*q0.z + p0.w*q0.w
             + p1.x*q1.x + p1.y*q1.y + p1.z*q1.z + p1.w*q1.w;
    aE[i] = (ee > 0.0f) ? ee : NEG_SLOPE * ee;
}

// ---------- 3: per-(dst,h) segment max ----------
__global__ __launch_bounds__(256) void segmax_k(const float* __restrict__ aE,
                                                const int* __restrict__ dst,
                                                unsigned* __restrict__ m_enc, int EH) {
    int i = blockIdx.x * 256 + threadIdx.x;
    if (i >= EH) return;
    int e = i >> 2, h = i & 3;
    atomicMax(&m_enc[(size_t)dst[e] * NHEAD + h], enc_f(aE[i]));
}

// ---------- 4: ex = exp(e - m[dst]); s[dst] += ex ----------
__global__ __launch_bounds__(256) void expsum_k(float* __restrict__ aE,
                                                const int* __restrict__ dst,
                                                const unsigned* __restrict__ m_enc,
                                                float* __restrict__ s, int EH) {
    int i = blockIdx.x * 256 + threadIdx.x;
    if (i >= EH) return;
    int e = i >> 2, h = i & 3;
    size_t nh = (size_t)dst[e] * NHEAD + h;
    float ex = __expf(aE[i] - dec_f(m_enc[nh]));
    aE[i] = ex;
    atomicAdd(&s[nh], ex);
}

// ---------- 5: a = ex / s[dst]; attn_sum[dst] += a ----------
__global__ __launch_bounds__(256) void norma_k(float* __restrict__ aE,
                                               const int* __restrict__ dst,
                                               const float* __restrict__ s,
                                               float* __restrict__ asum, int EH) {
    int i = blockIdx.x * 256 + threadIdx.x;
    if (i >= EH) return;
    int e = i >> 2, h = i & 3;
    size_t nh = (size_t)dst[e] * NHEAD + h;
    float a = aE[i] / s[nh];
    aE[i] = a;
    atomicAdd(&asum[nh], a);
}

// ---------- 6: out[dst] += a * ft[src]  (dominant kernel: ~3.3GB gather+scatter) ----------
__global__ __launch_bounds__(256) void msg_k(const float* __restrict__ ft,
                                             const float* __restrict__ aE,
                                             const int* __restrict__ src,
                                             const int* __restrict__ dst,
                                             float* __restrict__ out, long long total) {
    long long idx = (long long)blockIdx.x * 256 + threadIdx.x;
    if (idx >= total) return;
    int e = (int)(idx >> 8);       // /256 channels
    int c = (int)(idx & 255);      // h*64+d
    int h = c >> 6;
    float a = aE[(size_t)e * NHEAD + h];
    float v = a * ft[(size_t)src[e] * HD + c];      // coalesced 1KB gather per edge
    atomicAdd(out + (size_t)dst[e] * HD + c, v);    // coalesced atomic scatter
}

// ---------- 7: rst = out / (attn_sum + 1e-12) ----------
__global__ __launch_bounds__(256) void final_k(float* __restrict__ out,
                                               const float* __restrict__ asum,
                                               long long nHD) {
    long long i = (long long)blockIdx.x * 256 + threadIdx.x;
    if (i >= nHD) return;
    out[i] = out[i] / (asum[i >> 6] + 1e-12f);
}

extern "C" void kernel_launch(void* const* d_in, const int* in_sizes, int n_in,
                              void* d_out, int out_size, void* d_ws, size_t ws_size,
                              hipStream_t stream) {
    const float* feat   = (const float*)d_in[0];
    const float* e_feat = (const float*)d_in[1];
    const float* W      = (const float*)d_in[2];
    const float* attn_e = (const float*)d_in[3];
    const int*   src    = (const int*)d_in[4];
    const int*   dst    = (const int*)d_in[5];

    const int N = in_sizes[0] / IN_F;
    const int E = in_sizes[4];

    float* out = (float*)d_out;                  // [N, H*D] accumulator then rst
    float* aE  = out + (size_t)N * HD;           // [E, H]  e -> ex -> a (in place)

    float*    ft    = (float*)d_ws;                          // [N, 256]
    unsigned* m_enc = (unsigned*)(ft + (size_t)N * HD);      // [N, H]
    float*    s     = (float*)(m_enc + (size_t)N * NHEAD);   // [N, H]
    float*    asum  = s + (size_t)N * NHEAD;                 // [N, H]

    const long long nHD = (long long)N * HD;
    const long long nH  = (long long)N * NHEAD;
    const int EH        = E * NHEAD;
    const long long totalMsg = (long long)E * HD;

    init_k<<<(unsigned)((nHD + 255) / 256), 256, 0, stream>>>(out, m_enc, s, asum, nHD, nH);
    gemm_ft_wmma<<<(N + 15) / 16, 256, 0, stream>>>(feat, W, ft, N);
    edge_logits_k<<<(EH + 255) / 256, 256, 0, stream>>>(e_feat, attn_e, aE, EH);
    segmax_k<<<(EH + 255) / 256, 256, 0, stream>>>(aE, dst, m_enc, EH);
    expsum_k<<<(EH + 255) / 256, 256, 0, stream>>>(aE, dst, m_enc, s, EH);
    norma_k<<<(EH + 255) / 256, 256, 0, stream>>>(aE, dst, s, asum, EH);
    msg_k<<<(unsigned)((totalMsg + 255) / 256), 256, 0, stream>>>(ft, aE, src, dst, out, totalMsg);
    final_k<<<(unsigned)((nHD + 255) / 256), 256, 0, stream>>>(out, asum, nHD);
}